// Dot_22926535426138
// MI455X (gfx1250) — compile-verified
//
#include <hip/hip_runtime.h>
#include <stdint.h>

// CDNA5 / gfx1250: wave32, WMMA 16x16. FP32 path via v_wmma_f32_16x16x4_f32.
// Staging uses GLOBAL_LOAD_ASYNC_TO_LDS_B128 (ASYNCcnt) instead of
// global_load -> VGPR -> ds_store round trips.
typedef float v2f __attribute__((ext_vector_type(2)));
typedef float v8f __attribute__((ext_vector_type(8)));

#define BATCH   64
#define CH      64      // K (channels)
#define FLT     128     // N (filters)
#define LDA     68      // 64 + 4 pad; 68 floats = 272 B (16B-multiple, B128-safe)
#define LDB     132     // 128 + 4 pad; 132 floats = 528 B (16B-multiple)
#define BSTR    524288  // x batch stride   = 2*64*64*64
#define SSTR    262144  // x real/imag str  = 64*64*64
#define OBSTR   262144  // out batch stride = 2*32*32*128
#define OSSTR   131072  // out r/i stride   = 32*32*128

__device__ __forceinline__ uint32_t lds_off(const void* p) {
    // generic->local: LDS byte offset is the low 32 bits of the flat address
    return (uint32_t)(uintptr_t)p;
}

__device__ __forceinline__ void async_g2l_b128(uint32_t lds_addr,
                                               const float* __restrict__ sbase,
                                               uint32_t voff_bytes) {
    // GVS mode: mem = SGPR64 + VGPR32; VDST VGPR holds LDS byte address.
    asm volatile("global_load_async_to_lds_b128 %0, %1, %2"
                 :: "v"(lds_addr), "v"(voff_bytes), "s"(sbase)
                 : "memory");
}

__global__ __launch_bounds__(256)
void cplx_pos_gemm(const float* __restrict__ x,
                   const float* __restrict__ Wr,
                   const float* __restrict__ Wi,
                   const float* __restrict__ br,
                   const float* __restrict__ bi,
                   float* __restrict__ out)
{
    __shared__ float sAr[BATCH * LDA];   // xr tile [b][c]
    __shared__ float sAi[BATCH * LDA];   // xi tile [b][c]
    __shared__ float sBr[CH * LDB];      // Wr tile [c][f]
    __shared__ float sBi[CH * LDB];      // Wi tile [c][f]

    const int tid  = threadIdx.x;
    const int lane = tid & 31;
    const int wv   = tid >> 5;

    const int bw  = blockIdx.x;          // cropped w: 0..31
    const int bh  = blockIdx.y;          // cropped h: 0..31
    const int pos = bh * 32 + bw;
    const int h   = bh + 16;             // R_MIN = 16
    const int w   = bw + 16;             // C_MIN = 16

    // ---- async-stage weights: pos-th contiguous 64x128 block of Wr/Wi ----
    {
        const float* gWr = Wr + (size_t)pos * CH * FLT;
        const float* gWi = Wi + (size_t)pos * CH * FLT;
        #pragma unroll
        for (int it = 0; it < 8; ++it) {              // 2048 x 16B / 256 thr
            int idx = tid + it * 256;
            int row = idx >> 5;                       // 32 float4 per 128-f row
            int col = (idx & 31) << 2;
            uint32_t goff = (uint32_t)idx << 4;       // bytes
            async_g2l_b128(lds_off(&sBr[row * LDB + col]), gWr, goff);
            async_g2l_b128(lds_off(&sBi[row * LDB + col]), gWi, goff);
        }
    }
    // ---- async-stage x rows: 64 batches x 64 ch, real & imag planes ----
    {
        const float* xb = x + (size_t)h * 4096 + (size_t)w * 64;
        #pragma unroll
        for (int it = 0; it < 4; ++it) {              // 1024 x 16B / 256 thr
            int idx = tid + it * 256;
            int b   = idx >> 4;                       // 16 float4 per row
            int cc  = (idx & 15) << 2;
            uint32_t goffR = ((uint32_t)b * BSTR + (uint32_t)cc) << 2;  // bytes
            uint32_t goffI = goffR + (SSTR << 2);
            async_g2l_b128(lds_off(&sAr[b * LDA + cc]), xb, goffR);
            async_g2l_b128(lds_off(&sAi[b * LDA + cc]), xb, goffI);
        }
    }
    asm volatile("s_wait_asynccnt 0x0" ::: "memory");  // all LDS writes landed
    __syncthreads();

    // ---- per-wave tile assignment: m-tile = wv&3, n-tiles = (wv>>2)*4 + 0..3
    const int mt = wv & 3;
    const int ng = (wv >> 2) << 2;

    v8f accR[4] = {};   // real accumulators (16x16 f32 each)
    v8f accI[4] = {};   // imag accumulators

    const int l15   = lane & 15;
    const int khalf = (lane >> 4) << 1;          // lanes 16-31 hold K+2,K+3
    const int arow  = (mt * 16 + l15) * LDA;

    #pragma unroll
    for (int k = 0; k < CH; k += 4) {
        // A fragments (ISA 7.12.2: 32-bit A 16x4 — lane=M, vgpr+half=K)
        v2f ar = *(const v2f*)&sAr[arow + k + khalf];
        v2f ai = *(const v2f*)&sAi[arow + k + khalf];
        v2f nai = -ai;                            // fold rr - ii into one chain
        #pragma unroll
        for (int j = 0; j < 4; ++j) {
            const int cb = (ng + j) * 16 + l15;   // B: lane=N, vgpr+half=K
            v2f brf, bif;
            brf.x = sBr[(k + khalf)     * LDB + cb];
            brf.y = sBr[(k + khalf + 1) * LDB + cb];
            bif.x = sBi[(k + khalf)     * LDB + cb];
            bif.y = sBi[(k + khalf + 1) * LDB + cb];
            // acc_r += xr*Wr + (-xi)*Wi ; acc_i += xr*Wi + xi*Wr
            accR[j] = __builtin_amdgcn_wmma_f32_16x16x4_f32(false, ar,  false, brf, (short)0, accR[j], false, false);
            accR[j] = __builtin_amdgcn_wmma_f32_16x16x4_f32(false, nai, false, bif, (short)0, accR[j], false, false);
            accI[j] = __builtin_amdgcn_wmma_f32_16x16x4_f32(false, ar,  false, bif, (short)0, accI[j], false, false);
            accI[j] = __builtin_amdgcn_wmma_f32_16x16x4_f32(false, ai,  false, brf, (short)0, accI[j], false, false);
        }
    }

    // ---- bias + ReLU + store (D layout: vgpr r -> M=r / r+8; lane -> N) ----
    float* outP = out + (size_t)pos * FLT;
    const int mhi = (lane >> 4) << 3;            // +8 rows for upper lane half
    #pragma unroll
    for (int j = 0; j < 4; ++j) {
        const int f   = (ng + j) * 16 + l15;
        const float vbr = br[f];
        const float vbi = bi[f];
        #pragma unroll
        for (int r = 0; r < 8; ++r) {
            const int b = mt * 16 + r + mhi;
            const size_t o = (size_t)b * OBSTR + f;
            float rv = accR[j][r] + vbr; rv = rv > 0.f ? rv : 0.f;
            float iv = accI[j][r] + vbi; iv = iv > 0.f ? iv : 0.f;
            outP[o]         = rv;
            outP[o + OSSTR] = iv;
        }
    }
}

extern "C" void kernel_launch(void* const* d_in, const int* in_sizes, int n_in,
                              void* d_out, int out_size, void* d_ws, size_t ws_size,
                              hipStream_t stream) {
    const float* x  = (const float*)d_in[0];
    const float* Wr = (const float*)d_in[1];
    const float* Wi = (const float*)d_in[2];
    const float* br = (const float*)d_in[3];
    const float* bi = (const float*)d_in[4];
    float* out = (float*)d_out;
    (void)in_sizes; (void)n_in; (void)out_size; (void)d_ws; (void)ws_size;

    dim3 grid(32, 32, 1);   // one workgroup per cropped (h,w) position
    dim3 block(256, 1, 1);  // 8 wave32 waves
    cplx_pos_gemm<<<grid, block, 0, stream>>>(x, Wr, Wi, br, bi, out);
}